// CoverageLoss_81123342287506
// MI455X (gfx1250) — compile-verified
//
#include <hip/hip_runtime.h>
#include <hip/hip_bf16.h>
#include <math.h>

// Problem constants (fixed by the reference)
constexpr int N = 4096;   // rows
constexpr int D = 512;    // decoded dim
constexpr int L = 128;    // latent dim
constexpr int TOPK = 16;

typedef float v2f __attribute__((ext_vector_type(2)));
typedef float v8f __attribute__((ext_vector_type(8)));

// ---------------------------------------------------------------------------
// Kernel 1: C = A[ N x L ] @ W[ L x D ] + b, via V_WMMA_F32_16X16X4_F32.
// One wave per 16x16 output tile; K-loop of 32 chained WMMAs (compiler inserts
// the required hazard NOPs). blockIdx.y selects test (0) / train (1) latents.
// Fragment layouts per CDNA5 ISA 7.12.2:
//   A 16x4 f32 : lanes 0-15 -> K={k0,k0+1}, lanes 16-31 -> K={k0+2,k0+3}, M=lane%16
//   B 4x16 f32 : vgpr j holds row K = k0 + 2*half + j, N = lane%16
//   C/D 16x16  : vgpr v -> M = v + 8*half, N = lane%16
// ---------------------------------------------------------------------------
__global__ __launch_bounds__(32)
void gemm_bias_wmma(const float* __restrict__ testL, const float* __restrict__ trainL,
                    const float* __restrict__ W, const float* __restrict__ bias,
                    float* __restrict__ recTest, float* __restrict__ recTrain) {
  const float* A = (blockIdx.y == 0) ? testL : trainL;
  float* C       = (blockIdx.y == 0) ? recTest : recTrain;

  const int tile = blockIdx.x;            // (N/16)*(D/16) tiles, m-major
  const int tm = tile % (N / 16);
  const int tn = tile / (N / 16);
  const int m0 = tm * 16, n0 = tn * 16;

  const int lane = threadIdx.x;
  const int half = lane >> 4;             // 0: K lanes {0,1}, 1: K lanes {2,3}
  const int lm   = lane & 15;

  const float bv = bias[n0 + lm];
  v8f c;
  #pragma unroll
  for (int i = 0; i < 8; ++i) c[i] = bv;  // bias depends only on N(column)

  #pragma unroll 4
  for (int k0 = 0; k0 < L; k0 += 4) {
    // A fragment: two consecutive K values -> single 8B load
    v2f a = *(const v2f*)&A[(size_t)(m0 + lm) * L + k0 + 2 * half];
    // B fragment: W rows k0+2*half and k0+2*half+1, column n0+lm
    v2f w;
    w[0] = W[(size_t)(k0 + 2 * half + 0) * D + n0 + lm];
    w[1] = W[(size_t)(k0 + 2 * half + 1) * D + n0 + lm];
    c = __builtin_amdgcn_wmma_f32_16x16x4_f32(
        /*neg_a=*/false, a, /*neg_b=*/false, w,
        /*c_mod=*/(short)0, c, /*reuse_a=*/false, /*reuse_b=*/false);
  }

  #pragma unroll
  for (int v = 0; v < 8; ++v)
    C[(size_t)(m0 + v + 8 * half) * D + n0 + lm] = c[v];
}

// ---------------------------------------------------------------------------
// Kernel 2: fused L1-cdist + bottom-16 distances per row + score = mean(1/d).
// Block = 256 threads (8 waves). RM=16 rows per block (A rows cached whole in
// LDS, 32KB), RN=64 cols per tile, K streamed in 64-wide LDS chunks.
// Thread t: row ty=t>>4, col set {tx + 16*cc}, tx=t&15, cc=0..3.
// B LDS row stride 68 floats -> lane bank offset 4*tx (mod 64): conflict-free
// ds_load_b128. blockIdx.y: 0 = vs gt_vals (pos), 1 = vs recTrain (neg).
// ---------------------------------------------------------------------------
constexpr int RM = 16;
constexpr int RN = 64;
constexpr int BPAD = 68;   // floats per LDS B row (64 + 4), keeps 16B alignment

__global__ __launch_bounds__(256)
void cdist_topk(const float* __restrict__ recTest,
                const float* __restrict__ gt_vals,
                const float* __restrict__ recTrain,
                float* __restrict__ posS, float* __restrict__ negS) {
  __shared__ __align__(16) float sA[RM * D];        // 32 KB
  __shared__ __align__(16) float sB[RN * BPAD];     // 17 KB (reused as merge buf)

  const int t  = threadIdx.x;
  const int ty = t >> 4;       // row within block: 0..15
  const int tx = t & 15;       // column lane: 0..15
  const int row0 = blockIdx.x * RM;
  const float* __restrict__ B = blockIdx.y ? recTrain : gt_vals;
  float* __restrict__ outS    = blockIdx.y ? negS : posS;

  // Stage A rows into LDS (float4, fully coalesced)
  #pragma unroll
  for (int j = 0; j < (RM * D / 4) / 256; ++j) {
    int idx4 = t + j * 256;            // 0..2047
    int r    = idx4 >> 7;              // 0..15 (128 float4 per row)
    int k4   = idx4 & 127;
    *(float4*)&sA[r * D + k4 * 4] =
        *(const float4*)&recTest[(size_t)(row0 + r) * D + k4 * 4];
  }
  __syncthreads();

  const float INF = __builtin_huge_valf();
  float topv[TOPK];                    // register-resident bottom-16 distances
  #pragma unroll
  for (int i = 0; i < TOPK; ++i) topv[i] = INF;
  float mx = INF;                      // current max of topv

  for (int cb = 0; cb < N; cb += RN) {
    float acc[4] = {0.f, 0.f, 0.f, 0.f};
    for (int kt = 0; kt < D; kt += 64) {
      // Stage B chunk: 64 cols x 64 dims
      #pragma unroll
      for (int j = 0; j < 4; ++j) {
        int idx4 = t + j * 256;        // 0..1023
        int col  = idx4 >> 4;          // 0..63
        int k4   = idx4 & 15;
        *(float4*)&sB[col * BPAD + k4 * 4] =
            *(const float4*)&B[(size_t)(cb + col) * D + kt + k4 * 4];
      }
      __syncthreads();
      #pragma unroll 4
      for (int k4 = 0; k4 < 16; ++k4) {
        float4 av = *(const float4*)&sA[ty * D + kt + k4 * 4];   // 16-lane broadcast
        #pragma unroll
        for (int cc = 0; cc < 4; ++cc) {
          float4 bv = *(const float4*)&sB[(tx + cc * 16) * BPAD + k4 * 4];
          acc[cc] += fabsf(av.x - bv.x) + fabsf(av.y - bv.y)
                   + fabsf(av.z - bv.z) + fabsf(av.w - bv.w);
        }
      }
      __syncthreads();
    }
    // Insert the 4 finished distances into the per-thread bottom-16
    #pragma unroll
    for (int cc = 0; cc < 4; ++cc) {
      float d = acc[cc];
      if (d < mx) {
        bool done = false;
        #pragma unroll
        for (int i = 0; i < TOPK; ++i) {         // replace one instance of mx
          bool h = (!done) && (topv[i] == mx);
          if (h) topv[i] = d;
          done = done || h;
        }
        mx = topv[0];
        #pragma unroll
        for (int i = 1; i < TOPK; ++i) mx = fmaxf(mx, topv[i]);
      }
    }
  }

  // Sort local 16 ascending (unrolled insertion network, 225 comparators)
  #pragma unroll
  for (int i = 1; i < TOPK; ++i) {
    #pragma unroll
    for (int j = TOPK - 1; j > 0; --j) {
      float lo = fminf(topv[j - 1], topv[j]);
      float hi = fmaxf(topv[j - 1], topv[j]);
      topv[j - 1] = lo; topv[j] = hi;
    }
  }

  __syncthreads();                      // sB no longer needed -> overlay merge buf
  float* sMerge = sB;                   // 16 rows x 256 candidates (4096 <= 4352)
  #pragma unroll
  for (int i = 0; i < TOPK; ++i)
    sMerge[ty * 256 + tx * TOPK + i] = topv[i];
  __syncthreads();

  // One thread per row: 16-way tournament over 16 sorted lists.
  if (tx == 0) {
    float h[16]; int cn[16];
    #pragma unroll
    for (int j = 0; j < 16; ++j) { h[j] = sMerge[ty * 256 + j * TOPK]; cn[j] = 1; }
    float s = 0.f;
    for (int round = 0; round < TOPK; ++round) {
      float mn = h[0]; int mi = 0;
      #pragma unroll
      for (int j = 1; j < 16; ++j) if (h[j] < mn) { mn = h[j]; mi = j; }
      s += 1.0f / mn;                   // score = 1/distance (monotone)
      #pragma unroll
      for (int j = 0; j < 16; ++j) if (j == mi) {
        h[j] = (cn[j] < TOPK) ? sMerge[ty * 256 + j * TOPK + cn[j]] : INF;
        cn[j]++;
      }
    }
    outS[row0 + ty] = s * (1.0f / (float)TOPK);
  }
}

// ---------------------------------------------------------------------------
// Kernel 3: losses = relu(neg - pos); Huber(delta=1) vs 0; mean over N.
// ---------------------------------------------------------------------------
__global__ __launch_bounds__(256)
void finalize_loss(const float* __restrict__ posS, const float* __restrict__ negS,
                   float* __restrict__ out) {
  __shared__ float red[256];
  float s = 0.f;
  for (int i = threadIdx.x; i < N; i += 256) {
    float l = fmaxf(negS[i] - posS[i], 0.f);
    s += (l <= 1.0f) ? (0.5f * l * l) : (l - 0.5f);
  }
  red[threadIdx.x] = s;
  __syncthreads();
  for (int off = 128; off > 0; off >>= 1) {
    if (threadIdx.x < off) red[threadIdx.x] += red[threadIdx.x + off];
    __syncthreads();
  }
  if (threadIdx.x == 0) out[0] = red[0] * (1.0f / (float)N);
}

// ---------------------------------------------------------------------------
extern "C" void kernel_launch(void* const* d_in, const int* in_sizes, int n_in,
                              void* d_out, int out_size, void* d_ws, size_t ws_size,
                              hipStream_t stream) {
  const float* gt_vals = (const float*)d_in[0];   // [N, D]
  const float* trainL  = (const float*)d_in[1];   // [N, L]
  const float* testL   = (const float*)d_in[2];   // [N, L]
  const float* W       = (const float*)d_in[3];   // [L, D]
  const float* bias    = (const float*)d_in[4];   // [D]

  float* ws       = (float*)d_ws;
  float* recTest  = ws;                                   // N*D
  float* recTrain = recTest + (size_t)N * D;              // N*D
  float* posS     = recTrain + (size_t)N * D;             // N
  float* negS     = posS + N;                             // N

  // 1) Both decoder GEMMs (WMMA f32), grid.y selects test/train
  dim3 g1((N / 16) * (D / 16), 2);
  gemm_bias_wmma<<<g1, 32, 0, stream>>>(testL, trainL, W, bias, recTest, recTrain);

  // 2) Fused L1-cdist + bottom-16 + score, grid.y selects pos/neg matrix
  dim3 g2(N / RM, 2);
  cdist_topk<<<g2, 256, 0, stream>>>(recTest, gt_vals, recTrain, posS, negS);

  // 3) Huber mean -> scalar
  finalize_loss<<<1, 256, 0, stream>>>(posS, negS, (float*)d_out);
}